// LocalGCNLayer_21492016349940
// MI455X (gfx1250) — compile-verified
//
#include <hip/hip_runtime.h>
#include <math.h>

typedef __attribute__((ext_vector_type(2))) float v2f;
typedef __attribute__((ext_vector_type(8))) float v8f;

#define DFEAT 128
#define LN_EPSF 1e-5f

// ---------------- zero workspace ----------------
__global__ void k_zero(float* __restrict__ p, int n) {
  int stride = gridDim.x * blockDim.x;
  for (int i = blockIdx.x * blockDim.x + threadIdx.x; i < n; i += stride) p[i] = 0.0f;
}

// ---------------- degree by dst ----------------
__global__ void k_degree(const long long* __restrict__ ei, float* __restrict__ deg, int E) {
  int e = blockIdx.x * blockDim.x + threadIdx.x;
  if (e < E) {
    int dst = (int)ei[(long long)E + e];
    unsafeAtomicAdd(&deg[dst], 1.0f);
  }
}

// ---------------- dis = rsqrt(deg + 1) ----------------
__global__ void k_dis(const float* __restrict__ deg, float* __restrict__ dis, int n) {
  int i = blockIdx.x * blockDim.x + threadIdx.x;
  if (i < n) dis[i] = rsqrtf(deg[i] + 1.0f);
}

// ---------------- gather -> scale -> scatter-add (one wave32 per edge) ----------------
__global__ void k_scatter(const float* __restrict__ x, const long long* __restrict__ ei,
                          const float* __restrict__ ew, const float* __restrict__ dis,
                          float* __restrict__ agg, int E) {
  int wave = blockIdx.x * (blockDim.x >> 5) + (threadIdx.x >> 5);
  int lane = threadIdx.x & 31;
  if (wave >= E) return;
  long long src = ei[wave];
  long long dst = ei[(long long)E + wave];
  float s = ew[wave] * dis[src];
  // 128 floats per row: 32 lanes x float4 -> global_load_b128
  float4 v = reinterpret_cast<const float4*>(x + src * DFEAT)[lane];
  float* ap = agg + dst * DFEAT + lane * 4;
  unsafeAtomicAdd(ap + 0, v.x * s);
  unsafeAtomicAdd(ap + 1, v.y * s);
  unsafeAtomicAdd(ap + 2, v.z * s);
  unsafeAtomicAdd(ap + 3, v.w * s);
}

__device__ __forceinline__ float gelu_exact(float v) {
  return 0.5f * v * (1.0f + erff(v * 0.70710678118654752f));
}

// ---------------- fused: A = agg*dis + x ; h = gelu(A @ W^T + b) ; LN(h) ----------------
// block = 256 threads (8 waves). Each wave computes a 16x128 output tile via
// 32 k-steps x 8 n-tiles of v_wmma_f32_16x16x4_f32. W staged in XOR-swizzled LDS (64KB).
__global__ __launch_bounds__(256) void k_tail(
    const float* __restrict__ x, const float* __restrict__ agg, const float* __restrict__ dis,
    const float* __restrict__ w, const float* __restrict__ bias,
    const float* __restrict__ gamma, const float* __restrict__ beta,
    float* __restrict__ out, int n) {
  __shared__ float wl[DFEAT * DFEAT];  // wl[j][k ^ ((j&15)<<2)] = W[j][k]

  int tid = threadIdx.x;
  for (int idx = tid; idx < DFEAT * DFEAT; idx += 256) {
    int j = idx >> 7;
    int k = idx & 127;
    wl[(j << 7) + (k ^ ((j & 15) << 2))] = w[idx];
  }
  __syncthreads();

  int wv   = tid >> 5;
  int lane = tid & 31;
  int half = lane >> 4;   // 0: lanes 0-15 (K=k0,k0+1), 1: lanes 16-31 (K=k0+2,k0+3)
  int nl   = lane & 15;
  int row0 = blockIdx.x * 128 + wv * 16;

  int row  = row0 + nl;                 // A-matrix row this lane feeds
  int rowA = row < n ? row : (n - 1);   // clamp (stores are guarded)
  float disr = dis[rowA];
  const float* aggp = agg + (long long)rowA * DFEAT;
  const float* xp   = x   + (long long)rowA * DFEAT;
  __builtin_prefetch(aggp, 0, 0);       // global_prefetch_b8
  __builtin_prefetch(xp, 0, 0);

  v8f acc[8];
#pragma unroll
  for (int nt = 0; nt < 8; ++nt) acc[nt] = (v8f){0.f, 0.f, 0.f, 0.f, 0.f, 0.f, 0.f, 0.f};

  int swz = nl << 2;
#pragma unroll 4
  for (int ks = 0; ks < 32; ++ks) {
    int k0 = (ks << 2) + (half << 1);
    // A fragment: A[M=nl][k0..k0+1], fused residual A = agg*dis + x
    v2f av = *reinterpret_cast<const v2f*>(aggp + k0);
    v2f xv = *reinterpret_cast<const v2f*>(xp + k0);
    v2f a = av * disr + xv;
    int kidx = k0 ^ swz;  // swizzled, stays even -> 8B aligned
#pragma unroll
    for (int nt = 0; nt < 8; ++nt) {
      // B fragment: B[k0..k0+1][N=nl] = W[nt*16+nl][k0..k0+1]
      const v2f* bp = reinterpret_cast<const v2f*>(&wl[(((nt << 4) + nl) << 7) + kidx]);
      acc[nt] = __builtin_amdgcn_wmma_f32_16x16x4_f32(
          /*neg_a=*/false, a, /*neg_b=*/false, *bp,
          /*c_mod=*/(short)0, acc[nt], /*reuse_a=*/false, /*reuse_b=*/false);
    }
  }

  // per-column constants for this lane
  float bcol[8], gcol[8], btcol[8];
#pragma unroll
  for (int nt = 0; nt < 8; ++nt) {
    int col = (nt << 4) + nl;
    bcol[nt]  = bias[col];
    gcol[nt]  = gamma[col];
    btcol[nt] = beta[col];
  }

  // bias + exact GELU, accumulate row sums (rows m = r + 8*half live in vgpr r)
  float s[8], q[8];
#pragma unroll
  for (int r = 0; r < 8; ++r) { s[r] = 0.f; q[r] = 0.f; }
#pragma unroll
  for (int nt = 0; nt < 8; ++nt) {
#pragma unroll
    for (int r = 0; r < 8; ++r) {
      float v = acc[nt][r] + bcol[nt];
      v = gelu_exact(v);
      acc[nt][r] = v;
      s[r] += v;
      q[r] += v * v;
    }
  }
  // reduce across the 16 lanes of each half-wave (masks < 16 never cross halves)
#pragma unroll
  for (int m = 1; m < 16; m <<= 1) {
#pragma unroll
    for (int r = 0; r < 8; ++r) {
      s[r] += __shfl_xor(s[r], m, 32);
      q[r] += __shfl_xor(q[r], m, 32);
    }
  }
  float mu[8], rstd[8];
#pragma unroll
  for (int r = 0; r < 8; ++r) {
    float m_ = s[r] * (1.0f / 128.0f);
    float var = q[r] * (1.0f / 128.0f) - m_ * m_;
    mu[r] = m_;
    rstd[r] = rsqrtf(var + LN_EPSF);
  }

  // normalize + affine + store (row = row0 + r + 8*half, col = nt*16 + nl)
#pragma unroll
  for (int nt = 0; nt < 8; ++nt) {
#pragma unroll
    for (int r = 0; r < 8; ++r) {
      int grow = row0 + r + (half << 3);
      if (grow < n) {
        float o = (acc[nt][r] - mu[r]) * rstd[r] * gcol[nt] + btcol[nt];
        out[(long long)grow * DFEAT + (nt << 4) + nl] = o;
      }
    }
  }
}

extern "C" void kernel_launch(void* const* d_in, const int* in_sizes, int n_in,
                              void* d_out, int out_size, void* d_ws, size_t ws_size,
                              hipStream_t stream) {
  (void)n_in; (void)out_size; (void)ws_size;
  const float*     x     = (const float*)d_in[0];
  const long long* ei    = (const long long*)d_in[1];  // int64 edge_index, shape (2,E)
  const float*     ew    = (const float*)d_in[2];
  const float*     w     = (const float*)d_in[3];
  const float*     bias  = (const float*)d_in[4];
  const float*     gamma = (const float*)d_in[5];
  const float*     beta  = (const float*)d_in[6];

  int N = in_sizes[0] / DFEAT;
  int E = in_sizes[1] / 2;

  float* deg = (float*)d_ws;       // N floats
  float* dis = deg + N;            // N floats
  float* agg = dis + N;            // N*DFEAT floats

  int zcount = N * (DFEAT + 2);    // deg + dis + agg (dis overwritten below)
  k_zero<<<1024, 256, 0, stream>>>((float*)d_ws, zcount);
  k_degree<<<(E + 255) / 256, 256, 0, stream>>>(ei, deg, E);
  k_dis<<<(N + 255) / 256, 256, 0, stream>>>(deg, dis, N);
  k_scatter<<<(E + 7) / 8, 256, 0, stream>>>(x, ei, ew, dis, agg, E);
  k_tail<<<(N + 127) / 128, 256, 0, stream>>>(x, agg, dis, w, bias, gamma, beta,
                                              (float*)d_out, N);
}